// HeteroGAT_37237366456458
// MI455X (gfx1250) — compile-verified
//
#include <hip/hip_runtime.h>
#include <hip/hip_bf16.h>
#include <math.h>

// ---------------------------------------------------------------------------
// HeteroGAT on MI455X (gfx1250, wave32).
// Memory-bound scatter workload; dense projections run on V_WMMA_F32_16X16X4_F32
// (native f32 tensor op -> exact vs the f32 reference). Attention-score terms are
// algebraically folded into the projection weights so hd / he are never
// materialized:  (h * a).sum(-1) == x @ (W @ a).
// ---------------------------------------------------------------------------

typedef float v2f __attribute__((ext_vector_type(2)));
typedef float v8f __attribute__((ext_vector_type(8)));

static inline int cdiv_i(int a, int b) { return (a + b - 1) / b; }

// ---- float <-> order-preserving unsigned (for atomicMax on floats) ---------
__device__ __forceinline__ unsigned f2ord(float f) {
  unsigned u = __float_as_uint(f);
  return (u & 0x80000000u) ? ~u : (u | 0x80000000u);
}
__device__ __forceinline__ float ord2f(unsigned u) {
  u = (u & 0x80000000u) ? ~u : (u & 0x7fffffffu);
  return __uint_as_float(u);
}

// ---------------------------------------------------------------------------
// Kernel 1: f32 WMMA GEMM,  Y[N x M] = X[N x K] @ W[K x M], K compile-time,
// M multiple of 16. One wave per 16x16 tile. Branchless index clamping keeps
// every global_load unconditional (no exec-mask churn around the WMMA chain).
// A layout (ISA 7.12.2, 32-bit A 16x4):  VGPR j, lanes 0-15: K=j, lanes 16-31: K=2+j, M=lane%16
// B layout (symmetric):                  VGPR j, lanes 0-15: K=j, lanes 16-31: K=2+j, N=lane%16
// D layout: VGPR r, lanes 0-15: M=r, lanes 16-31: M=r+8, N=lane%16
// ---------------------------------------------------------------------------
template <int K>
__global__ void gemm_f32_wmma(const float* __restrict__ X,
                              const float* __restrict__ W,
                              float* __restrict__ Y,
                              int N, int M) {
  const int wave = (blockIdx.x * blockDim.x + threadIdx.x) >> 5;
  const int lane = threadIdx.x & 31;
  const int tilesM = M >> 4;
  const int tm = wave % tilesM;
  const int tn = wave / tilesM;
  const int row0 = tn << 4;
  if (row0 >= N) return;                 // wave-uniform: EXEC stays all-ones
  const int half = lane >> 4;
  const int l = lane & 15;
  const int r = min(row0 + l, N - 1);    // branchless clamp: rows >= N feed only
                                         // D rows >= N, which are never stored
  const int col = (tm << 4) + l;
  const float* __restrict__ Xrow = X + (size_t)r * K;
  const float* __restrict__ Wcol = W + col;

  v8f acc = {};
  constexpr int KSTEPS = (K + 3) / 4;
#pragma unroll
  for (int s = 0; s < KSTEPS; ++s) {
    v2f a, b;
#pragma unroll
    for (int j = 0; j < 2; ++j) {
      const int k = s * 4 + half * 2 + j;
      const bool ok = (k < K);           // folds away when K % 4 == 0
      const int kc = ok ? k : 0;
      const float av = Xrow[kc];
      const float bv = Wcol[(size_t)kc * M];
      a[j] = ok ? av : 0.0f;
      b[j] = ok ? bv : 0.0f;
    }
    acc = __builtin_amdgcn_wmma_f32_16x16x4_f32(
        /*neg_a=*/false, a, /*neg_b=*/false, b,
        /*c_mod=*/(short)0, acc, /*reuse_a=*/false, /*reuse_b=*/false);
  }

  float* __restrict__ Yp = Y + (size_t)(row0 + half * 8) * M + col;
  if (row0 + 16 <= N) {                  // wave-uniform fast path (always taken
#pragma unroll                           // for our N: 200000/100000/64 % 16 == 0)
    for (int rr = 0; rr < 8; ++rr) Yp[(size_t)rr * M] = acc[rr];
  } else {
#pragma unroll
    for (int rr = 0; rr < 8; ++rr) {
      if (row0 + half * 8 + rr < N) Yp[(size_t)rr * M] = acc[rr];
    }
  }
}

// ---------------------------------------------------------------------------
// Kernel 2: build folded weight matrices + folded edge-attention vectors.
// Wcat_task [12 x 96]: cols 0-31 Ws_tt | 32-47 Wr_dt | 48-63 Wr_vt | 64-79 Wr_tt
//   | 80-81 fold(Wd_dt,ad_dt) | 82-83 fold(Wd_vt,ad_vt)
//   | 84-85 fold(Ws_tt,as_tt) | 86-87 fold(Ws_tt,ad_tt) | 88-95 zero
// Wcat_data [5 x 48]: 0-31 Ws_dt | 32-33 fold(Ws_dt,as_dt) | rest zero
// Wcat_dev  [12 x 48]: 0-31 Ws_vt | 32-33 fold(Ws_vt,as_vt) | rest zero
// aew[16]: dt(h*3+k)@0..5, vt(h*2+k)@6..9, tt(h)@10..11
// ---------------------------------------------------------------------------
__global__ void build_weights(
    const float* __restrict__ Ws_dt, const float* __restrict__ Wd_dt,
    const float* __restrict__ as_dt, const float* __restrict__ ad_dt,
    const float* __restrict__ We_dt, const float* __restrict__ ae_dt,
    const float* __restrict__ Wr_dt,
    const float* __restrict__ Ws_vt, const float* __restrict__ Wd_vt,
    const float* __restrict__ as_vt, const float* __restrict__ ad_vt,
    const float* __restrict__ We_vt, const float* __restrict__ ae_vt,
    const float* __restrict__ Wr_vt,
    const float* __restrict__ Ws_tt,
    const float* __restrict__ as_tt, const float* __restrict__ ad_tt,
    const float* __restrict__ We_tt, const float* __restrict__ ae_tt,
    const float* __restrict__ Wr_tt,
    float* __restrict__ Wcat_task, float* __restrict__ Wcat_data,
    float* __restrict__ Wcat_dev, float* __restrict__ aew) {
  const int t = blockIdx.x * blockDim.x + threadIdx.x;
  if (t < 12 * 96) {
    const int k = t / 96, c = t % 96;
    float v = 0.0f;
    if (c < 32) v = Ws_tt[k * 32 + c];
    else if (c < 48) v = Wr_dt[k * 16 + (c - 32)];
    else if (c < 64) v = Wr_vt[k * 16 + (c - 48)];
    else if (c < 80) v = Wr_tt[k * 16 + (c - 64)];
    else if (c < 88) {
      const int q = c - 80, h = q & 1, which = q >> 1;
      const float* W = (which == 0) ? Wd_dt : (which == 1) ? Wd_vt : Ws_tt;
      const float* a = (which == 0) ? ad_dt : (which == 1) ? ad_vt
                       : (which == 2) ? as_tt : ad_tt;
      float s = 0.0f;
      for (int c2 = 0; c2 < 16; ++c2) s += W[k * 32 + h * 16 + c2] * a[h * 16 + c2];
      v = s;
    }
    Wcat_task[t] = v;
    return;
  }
  int t2 = t - 12 * 96;
  if (t2 < 5 * 48) {
    const int k = t2 / 48, c = t2 % 48;
    float v = 0.0f;
    if (c < 32) v = Ws_dt[k * 32 + c];
    else if (c < 34) {
      const int h = c - 32;
      float s = 0.0f;
      for (int c2 = 0; c2 < 16; ++c2) s += Ws_dt[k * 32 + h * 16 + c2] * as_dt[h * 16 + c2];
      v = s;
    }
    Wcat_data[t2] = v;
    return;
  }
  t2 -= 5 * 48;
  if (t2 < 12 * 48) {
    const int k = t2 / 48, c = t2 % 48;
    float v = 0.0f;
    if (c < 32) v = Ws_vt[k * 32 + c];
    else if (c < 34) {
      const int h = c - 32;
      float s = 0.0f;
      for (int c2 = 0; c2 < 16; ++c2) s += Ws_vt[k * 32 + h * 16 + c2] * as_vt[h * 16 + c2];
      v = s;
    }
    Wcat_dev[t2] = v;
    return;
  }
  t2 -= 12 * 48;
  if (t2 < 12) {
    float s = 0.0f;
    if (t2 < 6) {
      const int h = t2 / 3, k = t2 % 3;
      for (int c2 = 0; c2 < 16; ++c2) s += We_dt[k * 32 + h * 16 + c2] * ae_dt[h * 16 + c2];
    } else if (t2 < 10) {
      const int q = t2 - 6, h = q / 2, k = q % 2;
      for (int c2 = 0; c2 < 16; ++c2) s += We_vt[k * 32 + h * 16 + c2] * ae_vt[h * 16 + c2];
    } else {
      const int h = t2 - 10;
      for (int c2 = 0; c2 < 16; ++c2) s += We_tt[h * 16 + c2] * ae_tt[h * 16 + c2];
    }
    aew[t2] = s;
  }
}

// ---------------------------------------------------------------------------
// Small utility kernels
// ---------------------------------------------------------------------------
__global__ void amax_init(unsigned* __restrict__ a, int n) {
  const int i = blockIdx.x * blockDim.x + threadIdx.x;
  if (i < n) a[i] = 0x007fffffu;  // f2ord(-inf)
}
__global__ void iota_kernel(int* __restrict__ a, int n) {
  const int i = blockIdx.x * blockDim.x + threadIdx.x;
  if (i < n) a[i] = i;
}
__global__ void loop_stats(const int* __restrict__ dst, const float* __restrict__ eattr,
                           float* __restrict__ cnt, float* __restrict__ sm, int E) {
  const int e = blockIdx.x * blockDim.x + threadIdx.x;
  if (e >= E) return;
  const int d = dst[e];
  atomicAdd(&cnt[d], 1.0f);
  atomicAdd(&sm[d], eattr[e]);
}
__global__ void loop_attr_kernel(float* __restrict__ loopA, const float* __restrict__ cnt,
                                 const float* __restrict__ sm, int n) {
  const int i = blockIdx.x * blockDim.x + threadIdx.x;
  if (i < n) loopA[i] = sm[i] / fmaxf(cnt[i], 1.0f);
}

// ---------------------------------------------------------------------------
// Kernel 3: edge passes, templated on (DEDGE, PHASE) for straight-line codegen.
// PHASE 0: segment max; 1: softmax denominator; 2: weighted aggregation.
// ---------------------------------------------------------------------------
template <int DEDGE, int PHASE>
__global__ void edge_pass(int E,
                          const int* __restrict__ src, const int* __restrict__ dst,
                          const float* __restrict__ eattr, const float* __restrict__ aew,
                          const float* __restrict__ Hs, int ss, int s_soff,
                          const float* __restrict__ Hd, int ds, int d_soff,
                          unsigned* __restrict__ amax, float* __restrict__ den,
                          float* __restrict__ agg, int hs_coloff) {
  const int e = blockIdx.x * blockDim.x + threadIdx.x;
  if (e >= E) return;
  const int s = src[e], d = dst[e];
  float al0 = Hs[(size_t)s * ss + s_soff]     + Hd[(size_t)d * ds + d_soff];
  float al1 = Hs[(size_t)s * ss + s_soff + 1] + Hd[(size_t)d * ds + d_soff + 1];
#pragma unroll
  for (int k = 0; k < DEDGE; ++k) {
    const float v = eattr[(size_t)e * DEDGE + k];
    al0 += v * aew[k];
    al1 += v * aew[DEDGE + k];
  }
  al0 = (al0 > 0.0f) ? al0 : 0.2f * al0;   // leaky_relu(0.2)
  al1 = (al1 > 0.0f) ? al1 : 0.2f * al1;
  if (PHASE == 0) {
    atomicMax(&amax[d * 2 + 0], f2ord(al0));
    atomicMax(&amax[d * 2 + 1], f2ord(al1));
    return;
  }
  const float m0 = ord2f(amax[d * 2 + 0]);
  const float m1 = ord2f(amax[d * 2 + 1]);
  const float ex0 = __expf(al0 - m0);
  const float ex1 = __expf(al1 - m1);
  if (PHASE == 1) {
    atomicAdd(&den[d * 2 + 0], ex0);
    atomicAdd(&den[d * 2 + 1], ex1);
    return;
  }
  const float a0 = ex0 / (den[d * 2 + 0] + 1e-16f);
  const float a1 = ex1 / (den[d * 2 + 1] + 1e-16f);
  const float* hp = Hs + (size_t)s * ss + hs_coloff;
  float* ap = agg + (size_t)d * 16;
#pragma unroll
  for (int c = 0; c < 16; ++c)
    atomicAdd(&ap[c], hp[c] * a0 + hp[16 + c] * a1);
}

// ---------------------------------------------------------------------------
// Kernel 4: finalize. head-mean + residual + bias -> LayerNorm -> leaky(0.01),
// concat [x_tasks | ttf | dtf | vtf] into d_out [NT x 60].
// ---------------------------------------------------------------------------
__device__ __forceinline__ void rel_final(const float* __restrict__ agg_row,
                                          const float* __restrict__ resid_row,
                                          const float* __restrict__ b,
                                          const float* __restrict__ g,
                                          const float* __restrict__ be,
                                          float* __restrict__ o) {
  float v[16];
  float m = 0.0f;
#pragma unroll
  for (int c = 0; c < 16; ++c) {
    v[c] = agg_row[c] * 0.5f + resid_row[c] + b[c];  // mean over H=2 heads
    m += v[c];
  }
  m *= (1.0f / 16.0f);
  float var = 0.0f;
#pragma unroll
  for (int c = 0; c < 16; ++c) { const float dd = v[c] - m; var += dd * dd; }
  var *= (1.0f / 16.0f);
  const float inv = rsqrtf(var + 1e-5f);
#pragma unroll
  for (int c = 0; c < 16; ++c) {
    float y = (v[c] - m) * inv * g[c] + be[c];
    o[c] = (y > 0.0f) ? y : 0.01f * y;
  }
}

__global__ void finalize_kernel(const float* __restrict__ xt, const float* __restrict__ H,
                                const float* __restrict__ agg_dt, const float* __restrict__ agg_vt,
                                const float* __restrict__ agg_tt,
                                const float* __restrict__ b_dt, const float* __restrict__ g_dt,
                                const float* __restrict__ be_dt,
                                const float* __restrict__ b_vt, const float* __restrict__ g_vt,
                                const float* __restrict__ be_vt,
                                const float* __restrict__ b_tt, const float* __restrict__ g_tt,
                                const float* __restrict__ be_tt,
                                float* __restrict__ out, int NT) {
  const int n = blockIdx.x * blockDim.x + threadIdx.x;
  if (n >= NT) return;
  float* o = out + (size_t)n * 60;
  const float* x = xt + (size_t)n * 12;
#pragma unroll
  for (int j = 0; j < 12; ++j) o[j] = x[j];
  const float* hrow = H + (size_t)n * 96;
  rel_final(agg_tt + (size_t)n * 16, hrow + 64, b_tt, g_tt, be_tt, o + 12);
  rel_final(agg_dt + (size_t)n * 16, hrow + 32, b_dt, g_dt, be_dt, o + 28);
  rel_final(agg_vt + (size_t)n * 16, hrow + 48, b_vt, g_vt, be_vt, o + 44);
}

// ---------------------------------------------------------------------------
extern "C" void kernel_launch(void* const* d_in, const int* in_sizes, int n_in,
                              void* d_out, int out_size, void* d_ws, size_t ws_size,
                              hipStream_t stream) {
  const float* x_tasks  = (const float*)d_in[0];
  const float* x_data   = (const float*)d_in[1];
  const float* x_dev    = (const float*)d_in[2];
  const int*   e_dt_src = (const int*)d_in[3];
  const int*   e_dt_dst = (const int*)d_in[4];
  const int*   e_vt_src = (const int*)d_in[5];
  const int*   e_vt_dst = (const int*)d_in[6];
  const int*   e_tt_src = (const int*)d_in[7];
  const int*   e_tt_dst = (const int*)d_in[8];
  const float* e_dt_attr = (const float*)d_in[9];
  const float* e_vt_attr = (const float*)d_in[10];
  const float* e_tt_attr = (const float*)d_in[11];
  const float* Ws_dt = (const float*)d_in[12];
  const float* Wd_dt = (const float*)d_in[13];
  const float* as_dt = (const float*)d_in[14];
  const float* ad_dt = (const float*)d_in[15];
  const float* We_dt = (const float*)d_in[16];
  const float* ae_dt = (const float*)d_in[17];
  const float* Wr_dt = (const float*)d_in[18];
  const float* b_dt  = (const float*)d_in[19];
  const float* g_dt  = (const float*)d_in[20];
  const float* be_dt = (const float*)d_in[21];
  const float* Ws_vt = (const float*)d_in[22];
  const float* Wd_vt = (const float*)d_in[23];
  const float* as_vt = (const float*)d_in[24];
  const float* ad_vt = (const float*)d_in[25];
  const float* We_vt = (const float*)d_in[26];
  const float* ae_vt = (const float*)d_in[27];
  const float* Wr_vt = (const float*)d_in[28];
  const float* b_vt  = (const float*)d_in[29];
  const float* g_vt  = (const float*)d_in[30];
  const float* be_vt = (const float*)d_in[31];
  const float* Ws_tt = (const float*)d_in[32];
  const float* as_tt = (const float*)d_in[33];
  const float* ad_tt = (const float*)d_in[34];
  const float* We_tt = (const float*)d_in[35];
  const float* ae_tt = (const float*)d_in[36];
  const float* Wr_tt = (const float*)d_in[37];
  const float* b_tt  = (const float*)d_in[38];
  const float* g_tt  = (const float*)d_in[39];
  const float* be_tt = (const float*)d_in[40];

  const int NT  = in_sizes[0] / 12;
  const int ND  = in_sizes[1] / 5;
  const int NV  = in_sizes[2] / 12;
  const int EDT = in_sizes[3];
  const int EVT = in_sizes[5];
  const int ETT = in_sizes[7];

  // ---- workspace layout (float units) ----
  float* ws = (float*)d_ws;
  size_t o = 0;
  float* h_task = ws + o; o += (size_t)NT * 96;
  float* h_data = ws + o; o += (size_t)ND * 48;
  float* h_dev  = ws + o; o += (size_t)NV * 48;
  float* Wcat_t = ws + o; o += 12 * 96;
  float* Wcat_d = ws + o; o += 5 * 48;
  float* Wcat_v = ws + o; o += 12 * 48;
  float* aew    = ws + o; o += 16;
  int*   iotaA  = (int*)(ws + o); o += (size_t)NT;
  float* loopA  = ws + o; o += (size_t)NT;
  const size_t o_zero = o;                        // ---- zero-init region ----
  float* den    = ws + o; o += (size_t)NT * 6;    // 3 relations x [NT,2]
  float* agg    = ws + o; o += (size_t)NT * 48;   // 3 relations x [NT,16]
  float* cnt    = ws + o; o += (size_t)NT;
  float* smA    = ws + o; o += (size_t)NT;
  const size_t zero_bytes = (o - o_zero) * sizeof(float);
  unsigned* amax = (unsigned*)(ws + o); o += (size_t)NT * 6;

  float* den_dt = den;            float* den_vt = den + (size_t)NT * 2;  float* den_tt = den + (size_t)NT * 4;
  float* agg_dt = agg;            float* agg_vt = agg + (size_t)NT * 16; float* agg_tt = agg + (size_t)NT * 32;
  unsigned* amax_dt = amax;       unsigned* amax_vt = amax + (size_t)NT * 2;
  unsigned* amax_tt = amax + (size_t)NT * 4;

  const int B = 256;

  // ---- init ----
  hipMemsetAsync((char*)d_ws + o_zero * sizeof(float), 0, zero_bytes, stream);
  amax_init<<<cdiv_i(NT * 6, B), B, 0, stream>>>(amax, NT * 6);
  iota_kernel<<<cdiv_i(NT, B), B, 0, stream>>>(iotaA, NT);
  build_weights<<<8, B, 0, stream>>>(Ws_dt, Wd_dt, as_dt, ad_dt, We_dt, ae_dt, Wr_dt,
                                     Ws_vt, Wd_vt, as_vt, ad_vt, We_vt, ae_vt, Wr_vt,
                                     Ws_tt, as_tt, ad_tt, We_tt, ae_tt, Wr_tt,
                                     Wcat_t, Wcat_d, Wcat_v, aew);

  // ---- node projections via f32 WMMA ----
  {
    int tiles = cdiv_i(NT, 16) * (96 / 16);
    gemm_f32_wmma<12><<<cdiv_i(tiles * 32, B), B, 0, stream>>>(x_tasks, Wcat_t, h_task, NT, 96);
    tiles = cdiv_i(ND, 16) * (48 / 16);
    gemm_f32_wmma<5><<<cdiv_i(tiles * 32, B), B, 0, stream>>>(x_data, Wcat_d, h_data, ND, 48);
    tiles = cdiv_i(NV, 16) * (48 / 16);
    gemm_f32_wmma<12><<<cdiv_i(tiles * 32, B), B, 0, stream>>>(x_dev, Wcat_v, h_dev, NV, 48);
  }

  // ---- tt self-loop edge attr (scatter-mean by dst) ----
  loop_stats<<<cdiv_i(ETT, B), B, 0, stream>>>(e_tt_dst, e_tt_attr, cnt, smA, ETT);
  loop_attr_kernel<<<cdiv_i(NT, B), B, 0, stream>>>(loopA, cnt, smA, NT);

  // ---- edge passes: phase 0 (max), 1 (denom), 2 (aggregate) ----
#define EDGE_PHASE(PH)                                                                     \
  edge_pass<3, PH><<<cdiv_i(EDT, B), B, 0, stream>>>(EDT, e_dt_src, e_dt_dst, e_dt_attr,   \
      aew + 0, h_data, 48, 32, h_task, 96, 80, amax_dt, den_dt, agg_dt, 0);                \
  edge_pass<2, PH><<<cdiv_i(EVT, B), B, 0, stream>>>(EVT, e_vt_src, e_vt_dst, e_vt_attr,   \
      aew + 6, h_dev, 48, 32, h_task, 96, 82, amax_vt, den_vt, agg_vt, 0);                 \
  edge_pass<1, PH><<<cdiv_i(ETT, B), B, 0, stream>>>(ETT, e_tt_src, e_tt_dst, e_tt_attr,   \
      aew + 10, h_task, 96, 84, h_task, 96, 86, amax_tt, den_tt, agg_tt, 0);               \
  edge_pass<1, PH><<<cdiv_i(NT, B), B, 0, stream>>>(NT, iotaA, iotaA, loopA,               \
      aew + 10, h_task, 96, 84, h_task, 96, 86, amax_tt, den_tt, agg_tt, 0);
  EDGE_PHASE(0)
  EDGE_PHASE(1)
  EDGE_PHASE(2)
#undef EDGE_PHASE

  // ---- finalize ----
  finalize_kernel<<<cdiv_i(NT, B), B, 0, stream>>>(x_tasks, h_task, agg_dt, agg_vt, agg_tt,
      b_dt, g_dt, be_dt, b_vt, g_vt, be_vt, b_tt, g_tt, be_tt, (float*)d_out, NT);
}